// Word2VecModel_11785390260686
// MI455X (gfx1250) — compile-verified
//
#include <hip/hip_runtime.h>
#include <math.h>

typedef __attribute__((ext_vector_type(2))) float v2f;
typedef __attribute__((ext_vector_type(4))) float v4f;
typedef __attribute__((ext_vector_type(8))) float v8f;
typedef __attribute__((ext_vector_type(4))) int   v4i;

#define B_SZ 16384
#define C_SZ 10
#define K_SZ 20
#define D_SZ 128
#define ROWP 132   // padded LDS row stride (floats) to avoid bank conflicts

#if __has_builtin(__builtin_amdgcn_global_load_async_to_lds_b128) && \
    __has_builtin(__builtin_amdgcn_s_wait_asynccnt)
#define USE_ASYNC_LDS 1
typedef __attribute__((address_space(1))) v4i* gbl_v4i_p;   // global (prints as __device__)
typedef __attribute__((address_space(3))) v4i* lds_v4i_p;   // LDS    (prints as __shared__)
// 16B-per-lane direct global -> LDS copy, tracked on ASYNCcnt.
#define ASYNC_CP16(dst_lds, src_glb)                                   \
    __builtin_amdgcn_global_load_async_to_lds_b128(                    \
        (gbl_v4i_p)(src_glb), (lds_v4i_p)(dst_lds), 0, 0)
#else
#define USE_ASYNC_LDS 0
#endif

// One wave32 per batch element.
__global__ __launch_bounds__(32) void w2v_loss_kernel(
    const int* __restrict__ context,
    const int* __restrict__ target,
    const int* __restrict__ negs,
    const float* __restrict__ emb_W,
    const float* __restrict__ ctx_W,
    float* __restrict__ per_b_loss)
{
    __shared__ float sR[32][ROWP];   // rows 0..19: neg ctx rows, 20: target emb row, 21..31: zero
    __shared__ float sX[D_SZ];       // context mean vector
    __shared__ float sScore[32];

    const int b    = blockIdx.x;
    const int lane = threadIdx.x;        // 0..31
    const int hi   = (lane >= 16) ? 1 : 0;
    const int r    = lane & 15;

    // ---- 0) get the 21 gathered-row bases in flight early ----
    if (lane <= K_SZ) {
        const float* p = (lane < K_SZ)
            ? ctx_W + (size_t)negs[b * K_SZ + lane] * D_SZ
            : emb_W + (size_t)target[b] * D_SZ;
        __builtin_prefetch(p, 0, 0);         // row spans 4 cachelines;
        __builtin_prefetch(p + 64, 0, 0);    // touch front and middle
    }

#if USE_ASYNC_LDS
    // Issue the 21-row DMA into LDS *before* the context-mean phase so the
    // async engine overlaps it with the 10-row mean gather below.
    #pragma unroll
    for (int m = 0; m < K_SZ; ++m) {
        int idx = negs[b * K_SZ + m];
        ASYNC_CP16(&sR[m][lane * 4], ctx_W + (size_t)idx * D_SZ + lane * 4);
    }
    {
        int idx = target[b];
        ASYNC_CP16(&sR[K_SZ][lane * 4], emb_W + (size_t)idx * D_SZ + lane * 4);
    }
#endif

    // ---- 1) context mean: each lane owns a float4 slice of the 128-float row ----
    v4f acc = {0.f, 0.f, 0.f, 0.f};
    #pragma unroll
    for (int c = 0; c < C_SZ; ++c) {
        int idx = context[b * C_SZ + c];
        const v4f* row = reinterpret_cast<const v4f*>(ctx_W + (size_t)idx * D_SZ);
        acc += row[lane];
    }
    acc *= (1.0f / C_SZ);
    *reinterpret_cast<v4f*>(&sX[lane * 4]) = acc;

    // ---- 2) stage 21 gathered rows into LDS (full coalesced 512B rows) ----
#if !USE_ASYNC_LDS
    #pragma unroll
    for (int m = 0; m < K_SZ; ++m) {
        int idx = negs[b * K_SZ + m];
        const v4f* row = reinterpret_cast<const v4f*>(ctx_W + (size_t)idx * D_SZ);
        *reinterpret_cast<v4f*>(&sR[m][lane * 4]) = row[lane];
    }
    {
        int idx = target[b];
        const v4f* row = reinterpret_cast<const v4f*>(emb_W + (size_t)idx * D_SZ);
        *reinterpret_cast<v4f*>(&sR[K_SZ][lane * 4]) = row[lane];
    }
#endif
    {
        v4f z = {0.f, 0.f, 0.f, 0.f};
        #pragma unroll
        for (int m = K_SZ + 1; m < 32; ++m)
            *reinterpret_cast<v4f*>(&sR[m][lane * 4]) = z;
    }
#if USE_ASYNC_LDS
    __builtin_amdgcn_s_wait_asynccnt(0);
#endif
    __syncthreads();

    // ---- 3) scores[m] = sum_d R[m][d] * x[d] via V_WMMA_F32_16X16X4_F32 ----
    // A-matrix 16x4 f32 layout (ISA 7.12.2): lanes 0-15 M=0..15 {VGPR0=K0, VGPR1=K1},
    //                                        lanes 16-31 M=0..15 {VGPR0=K2, VGPR1=K3}.
    // B-matrix 4x16: rows striped across lanes within a VGPR; x broadcast over all N.
    v8f acc0 = {};
    v8f acc1 = {};
    #pragma unroll 4
    for (int kk = 0; kk < D_SZ; kk += 4) {
        v2f a0, a1, bb;
        a0.x = sR[r][kk + 2 * hi];
        a0.y = sR[r][kk + 1 + 2 * hi];
        a1.x = sR[16 + r][kk + 2 * hi];
        a1.y = sR[16 + r][kk + 1 + 2 * hi];
        bb.x = sX[kk + hi];
        bb.y = sX[kk + 2 + hi];
        acc0 = __builtin_amdgcn_wmma_f32_16x16x4_f32(
            false, a0, false, bb, (short)0, acc0, false, false);
        acc1 = __builtin_amdgcn_wmma_f32_16x16x4_f32(
            false, a1, false, bb, (short)0, acc1, false, false);
    }

    // ---- 4) extract column N=0 of D: lane 0 holds M=0..7, lane 16 holds M=8..15 ----
    if (r == 0) {
        #pragma unroll
        for (int j = 0; j < 8; ++j) {
            sScore[hi * 8 + j]      = acc0[j];   // tile 0: rows 0..15
            sScore[16 + hi * 8 + j] = acc1[j];   // tile 1: rows 16..31
        }
    }
    __syncthreads();

    // ---- 5) log-sigmoid terms + wave reduction ----
    float ls = 0.f;
    if (lane <= K_SZ) {
        float s = sScore[lane];
        float z = (lane == K_SZ) ? s : -s;              // pos uses +score, negs use -score
        ls = fminf(z, 0.f) - log1pf(__expf(-fabsf(z))); // stable log_sigmoid(z)
    }
    #pragma unroll
    for (int off = 16; off > 0; off >>= 1)
        ls += __shfl_down(ls, off, 32);
    if (lane == 0)
        per_b_loss[b] = -ls;
}

// Deterministic fixed-tree mean reduction of B_SZ floats.
__global__ __launch_bounds__(1024) void reduce_mean_kernel(
    const float* __restrict__ in, float* __restrict__ out)
{
    __shared__ float s[1024];
    const int t = threadIdx.x;
    float v = 0.f;
    for (int i = t; i < B_SZ; i += 1024)
        v += in[i];
    s[t] = v;
    __syncthreads();
    #pragma unroll
    for (int stride = 512; stride > 0; stride >>= 1) {
        if (t < stride) s[t] += s[t + stride];
        __syncthreads();
    }
    if (t == 0)
        out[0] = s[0] * (1.0f / B_SZ);
}

extern "C" void kernel_launch(void* const* d_in, const int* in_sizes, int n_in,
                              void* d_out, int out_size, void* d_ws, size_t ws_size,
                              hipStream_t stream) {
    const int*   context = (const int*)d_in[0];
    const int*   target  = (const int*)d_in[1];
    const int*   negs    = (const int*)d_in[2];
    const float* emb_W   = (const float*)d_in[3];
    const float* ctx_W   = (const float*)d_in[4];

    float* per_b = (float*)d_ws;   // B_SZ floats of scratch

    w2v_loss_kernel<<<B_SZ, 32, 0, stream>>>(context, target, negs, emb_W, ctx_W, per_b);
    reduce_mean_kernel<<<1, 1024, 0, stream>>>(per_b, (float*)d_out);
}